// Discriminator_26895085208120
// MI455X (gfx1250) — compile-verified
//
#include <hip/hip_runtime.h>

// out[b, i] = cos(x[b,i] + thetas[i])   for b in [0,32), i in [0,20)
//
// Derivation: all gates are RX on distinct wires (commute), the initial state
// |0..0> is a product state, and RX(a)RX(b) = RX(a+b) on the same wire, so
// qubit i of batch b ends in [cos(phi/2), -i sin(phi/2)] with
// phi = x[b,i] + thetas[i], giving <Z_i> = cos(phi). This replaces the
// 256 MB-statevector simulation (~20 GB HBM traffic) with 640 cos ops.
//
// The broadcast add x + thetas is performed on the CDNA5 matrix pipe as a
// rank-1 WMMA: D = A x B + C with A(16x4) = e0 column of ones,
// B(4x16) = thetas in row K=0, C = x tile. V_WMMA_F32_16X16X4_F32 keeps
// everything in f32 (numerically identical to an FMA). The rank-1 structure
// (only K=0 nonzero in both A and B) makes the result independent of which
// K-slice maps to which lane half.

typedef __attribute__((ext_vector_type(2))) float v2f;
typedef __attribute__((ext_vector_type(8))) float v8f;

__global__ __launch_bounds__(32)
void rx_expval_wmma(const float* __restrict__ x,
                    const float* __restrict__ thetas,
                    float* __restrict__ out,
                    int B, int NQ) {
    const int lane = threadIdx.x;      // wave32: 0..31, EXEC all ones
    const int m0   = blockIdx.x * 16;  // batch tile origin
    const int n0   = blockIdx.y * 16;  // qubit tile origin
    const int col  = lane & 15;        // N index within tile
    const int hi   = lane >> 4;        // lane half

    const int  q   = n0 + col;
    const bool qok = q < NQ;
    const int  qc  = qok ? q : 0;      // clamped column (safe address)

    // A matrix 16x4 f32: A[m][k] = (k==0) ? 1 : 0.
    // Lanes 0..15 hold row M=lane; their VGPR0 carries K=0 -> 1.0.
    v2f a;
    a.x = (hi == 0) ? 1.0f : 0.0f;
    a.y = 0.0f;

    // B matrix 4x16 f32: B[0][n] = thetas[n0+n], rows 1..3 zero.
    float th = thetas[qc];             // branchless load (address clamped)
    if (!(hi == 0 && qok)) th = 0.0f;  // keep only row K=0, valid columns
    v2f b;
    b.x = th;
    b.y = 0.0f;

    // C matrix 16x16 f32 accumulator layout:
    // VGPR r: lanes 0-15 -> (M=r, N=lane), lanes 16-31 -> (M=r+8, N=lane-16)
    v8f c;
#pragma unroll
    for (int r = 0; r < 8; ++r) {
        int  row   = m0 + r + hi * 8;
        bool rowok = row < B;
        int  rc    = rowok ? row : 0;  // clamped row (safe address)
        float v = x[rc * NQ + qc];     // unconditional load
        if (!(rowok && qok)) v = 0.0f; // v_cndmask, no exec divergence
        c[r] = v;
    }

    // D = ones_col * thetas_row + x   ->  D[b][i] = x[b,i] + thetas[i]
    v8f d = __builtin_amdgcn_wmma_f32_16x16x4_f32(
        /*neg_a=*/false, a, /*neg_b=*/false, b,
        /*c_mod=*/(short)0, c, /*reuse_a=*/false, /*reuse_b=*/false);

#pragma unroll
    for (int r = 0; r < 8; ++r) {
        int row = m0 + r + hi * 8;
        if (row < B && qok) out[row * NQ + q] = __builtin_cosf(d[r]);
    }
}

extern "C" void kernel_launch(void* const* d_in, const int* in_sizes, int n_in,
                              void* d_out, int out_size, void* d_ws, size_t ws_size,
                              hipStream_t stream) {
    const float* x      = (const float*)d_in[0];  // [B, NQ] f32
    const float* thetas = (const float*)d_in[1];  // [NQ]    f32
    // d_in[2] = n_qubits scalar — shapes recovered from in_sizes instead.
    const int NQ = in_sizes[1];                   // 20
    const int B  = in_sizes[0] / NQ;              // 32

    dim3 grid((B + 15) / 16, (NQ + 15) / 16);     // 2 x 2 tiles of 16x16
    rx_expval_wmma<<<grid, 32, 0, stream>>>(x, thetas, (float*)d_out, B, NQ);
}